// BiLstm_CRF_34050500723218
// MI455X (gfx1250) — compile-verified
//
#include <hip/hip_runtime.h>
#include <hip/hip_bf16.h>
#include <math.h>

typedef float v2f __attribute__((ext_vector_type(2)));
typedef float v8f __attribute__((ext_vector_type(8)));

#define TT 4096
#define E_DIM 256
#define H_DIM 256
#define G4 1024        // 4*H
#define NTAG 32
#define START_TAG 30
#define END_TAG 31
#define NEGV -10000.0f

#define NW 8             // workgroups per direction for the recurrence
#define HSL 32           // h elements per workgroup (H_DIM / NW)
#define ROWS 128         // gate rows per workgroup (4 * HSL)
#define WPAD 260         // padded LDS row stride (bank-conflict-free)

// ---------------- embedding gather: x[t] = emb[sent[t]] ----------------
__global__ void gather_emb(const int* __restrict__ sent,
                           const float* __restrict__ emb,
                           float* __restrict__ x) {
  int idx = blockIdx.x * blockDim.x + threadIdx.x;   // T*64 float4 slots
  int t  = idx >> 6;
  int e4 = idx & 63;
  const float4* es = (const float4*)emb;
  float4* xd = (float4*)x;
  xd[(size_t)t * 64 + e4] = es[(size_t)sent[t] * 64 + e4];
}

// ------------- fp32 WMMA GEMM: C[M,N] = A[M,K] * Bw[N,K]^T (+bias) -------------
// One wave computes a 16 x (16*NT) strip. Software-pipelined: loads for k+4 are
// issued before the WMMA burst for k, so waits land on deep loadcnt values.
template <int NT>
__global__ void gemm_wmma(const float* __restrict__ A,
                          const float* __restrict__ Bw,
                          const float* __restrict__ bias,   // [N] or nullptr
                          float* __restrict__ C,
                          int M, int N, int K, int lda, int ldb, int ldc) {
  int wavesPerBlock = blockDim.x >> 5;
  int waveId = blockIdx.x * wavesPerBlock + (threadIdx.x >> 5);
  int lane   = threadIdx.x & 31;
  int groupsN = (N >> 4) / NT;
  int tm  = (waveId / groupsN) << 4;
  int tn0 = (waveId % groupsN) * (NT * 16);
  if (tm >= M) return;

  int half = lane >> 4;     // 0: K base 0 ; 1: K base 2
  int mn   = lane & 15;     // A row / B col within tile

  const float* Arow = A + (size_t)(tm + mn) * lda + 2 * half;
  const float* Brow[NT];
  #pragma unroll
  for (int nt = 0; nt < NT; ++nt)
    Brow[nt] = Bw + (size_t)(tn0 + nt * 16 + mn) * ldb + 2 * half;

  v8f c[NT];
  #pragma unroll
  for (int nt = 0; nt < NT; ++nt) c[nt] = (v8f){};

  // pipeline prologue
  v2f a = *(const v2f*)(Arow);
  v2f b[NT];
  #pragma unroll
  for (int nt = 0; nt < NT; ++nt) b[nt] = *(const v2f*)(Brow[nt]);

  for (int k0 = 0; k0 < K - 4; k0 += 4) {
    v2f an = *(const v2f*)(Arow + k0 + 4);
    v2f bn[NT];
    #pragma unroll
    for (int nt = 0; nt < NT; ++nt) bn[nt] = *(const v2f*)(Brow[nt] + k0 + 4);
    #pragma unroll
    for (int nt = 0; nt < NT; ++nt)
      c[nt] = __builtin_amdgcn_wmma_f32_16x16x4_f32(
          false, a, false, b[nt], (short)0, c[nt], false, false);
    a = an;
    #pragma unroll
    for (int nt = 0; nt < NT; ++nt) b[nt] = bn[nt];
  }
  #pragma unroll
  for (int nt = 0; nt < NT; ++nt)
    c[nt] = __builtin_amdgcn_wmma_f32_16x16x4_f32(
        false, a, false, b[nt], (short)0, c[nt], false, false);

  #pragma unroll
  for (int nt = 0; nt < NT; ++nt) {
    float bb = bias ? bias[tn0 + nt * 16 + mn] : 0.0f;
    #pragma unroll
    for (int r = 0; r < 8; ++r)
      C[(size_t)(tm + r + 8 * half) * ldc + tn0 + nt * 16 + mn] = c[nt][r] + bb;
  }
}

// ---------------- flag reset (deterministic across graph replays) ----------------
__global__ void zero_flags(int* __restrict__ flags) {
  if (threadIdx.x < 2 * NW) flags[threadIdx.x] = 0;
}

// -------- sequential LSTM: 8 WGs per direction, Whh shard resident in LDS --------
// WG (dir,w) owns h[j], j in [w*32, w*32+32), and gate rows {j, 256+j, 512+j, 768+j}.
// Per step: LDS-only GEMV, tiny h-slice exchange through L2, flag spin-barrier.
// Static __shared__ (~135 KB) — CDNA5 allows up to 320 KB LDS per workgroup.
__global__ void __launch_bounds__(256)
lstm_mw(const float* __restrict__ xW,      // [T, 2048]: dir0 cols 0..1023, dir1 1024..2047
        const float* __restrict__ Whh_f, const float* __restrict__ b_f,
        const float* __restrict__ Whh_b, const float* __restrict__ b_b,
        const float* __restrict__ h0,    const float* __restrict__ c0,
        float* __restrict__ hbuf,        // [2][512] ping-pong h exchange
        int* __restrict__ flags,         // [16] monotonic step counters
        float* __restrict__ hcat) {      // [T, 512]
  __shared__ float w_lds[ROWS * WPAD];
  __shared__ float h_s[H_DIM];
  __shared__ float gate_s[ROWS];

  const int w   = blockIdx.x & (NW - 1);
  const int dir = blockIdx.x >> 3;
  const int tid = threadIdx.x;
  const float* Whh = dir ? Whh_b : Whh_f;
  const float* bv  = dir ? b_b   : b_f;

  const int lr    = tid >> 1;      // local gate row 0..127
  const int halfd = tid & 1;       // which half of the 256-dot
  const int gtype = lr >> 5;       // 0=i 1=f 2=g 3=o
  const int jl    = lr & 31;
  const int gr    = gtype * H_DIM + w * HSL + jl;   // global gate row

  // preload Whh shard into LDS (padded rows)
  for (int idx = tid; idx < ROWS * H_DIM; idx += 256) {
    int r = idx >> 8, e = idx & 255;
    int grr = (r >> 5) * H_DIM + w * HSL + (r & 31);
    w_lds[r * WPAD + e] = Whh[(size_t)grr * H_DIM + e];
  }
  if (tid < H_DIM) h_s[tid] = h0[dir * H_DIM + tid];
  float cst = 0.0f;
  if (tid < HSL) cst = c0[dir * H_DIM + w * HSL + tid];
  const float bg = bv[gr];
  __syncthreads();

  const int fbase = dir * NW;
  for (int s = 0; s < TT; ++s) {
    const int t = dir ? (TT - 1 - s) : s;
    if (halfd == 0 && s + 1 < TT) {
      int tn_ = dir ? (TT - 2 - s) : (s + 1);
      __builtin_prefetch(&xW[(size_t)tn_ * 2048 + dir * 1024 + gr], 0, 1);
    }

    // half-dot over LDS-resident weights
    const float4* wr = (const float4*)(w_lds + lr * WPAD + halfd * 128);
    const float4* hr = (const float4*)(h_s + halfd * 128);
    float acc = 0.0f;
    #pragma unroll
    for (int e = 0; e < 32; ++e) {
      float4 ww = wr[e];
      float4 hh = hr[e];
      acc += ww.x * hh.x + ww.y * hh.y + ww.z * hh.z + ww.w * hh.w;
    }
    acc += __shfl_xor(acc, 1, 32);
    if (halfd == 0)
      gate_s[lr] = acc + bg + xW[(size_t)t * 2048 + dir * 1024 + gr];
    __syncthreads();

    if (tid < HSL) {
      float iv = gate_s[tid];
      float fv = gate_s[HSL + tid];
      float gv = gate_s[2 * HSL + tid];
      float ov = gate_s[3 * HSL + tid];
      iv = 1.0f / (1.0f + __expf(-iv));
      fv = 1.0f / (1.0f + __expf(-fv));
      ov = 1.0f / (1.0f + __expf(-ov));
      cst = fv * cst + iv * tanhf(gv);
      float hv = ov * tanhf(cst);
      int j = w * HSL + tid;
      hcat[(size_t)t * (2 * H_DIM) + dir * H_DIM + j] = hv;
      hbuf[(s & 1) * 512 + dir * H_DIM + j] = hv;
    }
    __threadfence();
    __syncthreads();
    if (tid == 0)
      __hip_atomic_store(&flags[fbase + w], s + 1, __ATOMIC_RELEASE,
                         __HIP_MEMORY_SCOPE_AGENT);
    if (tid < NW) {
      while (__hip_atomic_load(&flags[fbase + tid], __ATOMIC_ACQUIRE,
                               __HIP_MEMORY_SCOPE_AGENT) < s + 1)
        __builtin_amdgcn_s_sleep(1);
    }
    __syncthreads();
    // refresh full h from the exchange buffer (device-scope loads bypass stale L0)
    if (tid < H_DIM)
      h_s[tid] = __hip_atomic_load(&hbuf[(s & 1) * 512 + dir * H_DIM + tid],
                                   __ATOMIC_RELAXED, __HIP_MEMORY_SCOPE_AGENT);
    __syncthreads();
  }
}

// ---------------- Viterbi: one wave32, lane == tag ----------------
__global__ void viterbi(const float* __restrict__ feats,   // [T, 32]
                        const float* __restrict__ crf,     // [32, 32]
                        int* __restrict__ steps,           // [T, 32]
                        float* __restrict__ out) {         // labels[T] + score
  int lane = threadIdx.x;
  __shared__ float crf_s[NTAG * NTAG];
  __shared__ float sp_s[NTAG];
  __shared__ float se_s[NTAG];

  for (int p = 0; p < NTAG; ++p) crf_s[p * NTAG + lane] = crf[p * NTAG + lane];
  float sp = (lane == START_TAG) ? 0.0f : NEGV;
  __syncthreads();

  for (int t = 0; t < TT; ++t) {
    sp_s[lane] = sp;
    __syncthreads();
    float feat = feats[t * NTAG + lane];
    float best = -3.4e38f;
    int arg = 0;
    #pragma unroll
    for (int p = 0; p < NTAG; ++p) {
      float v = sp_s[p] + crf_s[p * NTAG + lane];
      if (v > best) { best = v; arg = p; }
    }
    sp = best + feat;
    steps[t * NTAG + lane] = arg;
    __syncthreads();
  }

  se_s[lane] = sp + crf_s[lane * NTAG + END_TAG];
  __syncthreads();
  if (lane == 0) {
    float bestv = -3.4e38f;
    int best = 0;
    for (int p = 0; p < NTAG; ++p)
      if (se_s[p] > bestv) { bestv = se_s[p]; best = p; }
    int lbl = best;
    for (int t = TT - 1; t >= 0; --t) {
      out[t] = (float)lbl;
      lbl = steps[t * NTAG + lbl];
    }
    out[TT] = bestv;
  }
}

extern "C" void kernel_launch(void* const* d_in, const int* in_sizes, int n_in,
                              void* d_out, int out_size, void* d_ws, size_t ws_size,
                              hipStream_t stream) {
  const int*   sent  = (const int*)d_in[0];
  const float* emb   = (const float*)d_in[2];
  const float* Wih_f = (const float*)d_in[3];
  const float* Whh_f = (const float*)d_in[4];
  const float* b_f   = (const float*)d_in[5];
  const float* Wih_b = (const float*)d_in[6];
  const float* Whh_b = (const float*)d_in[7];
  const float* b_b   = (const float*)d_in[8];
  const float* Wout  = (const float*)d_in[9];
  const float* bout  = (const float*)d_in[10];
  const float* crf   = (const float*)d_in[11];
  const float* h0    = (const float*)d_in[12];
  const float* c0    = (const float*)d_in[13];
  float* out = (float*)d_out;

  char* ws = (char*)d_ws;
  float* x     = (float*)(ws);                               // 4 MB   [T,256]
  float* xW    = (float*)(ws + ((size_t)4  << 20));          // 32 MB  [T,2048]
  float* hcat  = (float*)(ws + ((size_t)36 << 20));          // 8 MB   [T,512]
  float* feats = (float*)(ws + ((size_t)44 << 20));          // 512 KB [T,32]
  int*   steps = (int*)  (ws + ((size_t)44 << 20) + ((size_t)512 << 10)); // 512 KB
  float* hbuf  = (float*)(ws + ((size_t)45 << 20));          // 4 KB
  int*   flags = (int*)  (ws + ((size_t)45 << 20) + 8192);   // 64 B

  // 1) embedding gather
  gather_emb<<<(TT * 64) / 256, 256, 0, stream>>>(sent, emb, x);

  // 2) input projections via fp32 WMMA: [4096,256]x[256,1024] per direction
  {
    int waves = (TT / 16) * ((G4 / 16) / 4);   // 16x64 strip per wave
    gemm_wmma<4><<<waves / 4, 128, 0, stream>>>(x, Wih_f, nullptr, xW,
                                                TT, G4, E_DIM, E_DIM, E_DIM, 2048);
    gemm_wmma<4><<<waves / 4, 128, 0, stream>>>(x, Wih_b, nullptr, xW + G4,
                                                TT, G4, E_DIM, E_DIM, E_DIM, 2048);
  }

  // 3) recurrences: 8 WGs/direction, Whh sharded into LDS, flag-synced steps
  zero_flags<<<1, 32, 0, stream>>>(flags);
  lstm_mw<<<2 * NW, 256, 0, stream>>>(xW, Whh_f, b_f, Whh_b, b_b,
                                      h0, c0, hbuf, flags, hcat);

  // 4) output projection via WMMA: [4096,512]x[512,32] + bout
  {
    int waves = (TT / 16) * ((NTAG / 16) / 2);  // 16x32 strip per wave
    gemm_wmma<2><<<waves / 4, 128, 0, stream>>>(hcat, Wout, bout, feats,
                                                TT, NTAG, 2 * H_DIM, 2 * H_DIM,
                                                2 * H_DIM, NTAG);
  }

  // 5) Viterbi DP + backtrack
  viterbi<<<1, 32, 0, stream>>>(feats, crf, steps, out);
}